// YoloLoss_41807211659944
// MI455X (gfx1250) — compile-verified
//
#include <hip/hip_runtime.h>
#include <math.h>

// ---------------------------------------------------------------------------
// YOLO loss for MI455X (gfx1250, wave32).
// HBM-bound problem (~80MB f32 inputs -> ~4us floor at 23.3 TB/s). Strategy:
//   k_count : per-(b,anchor,gt) positive counts -> n_pos / any_pos per image
//   k_topk  : fallback assignment. dist^2 is a rank-4 bilinear form, computed
//             16x16 anchors-x-gts per V_WMMA_F32_16X16X4_F32 issue (full f32
//             precision: A=(cx,cy,cx^2+cy^2,1), B=(-2gx,-2gy,1,|g|^2)).
//             Unsorted replace-max top-10; incremental (x,y) -> no div/mod
//             in the hot loop.
//   k_main  : one fused streaming pass per anchor (coalesced strided loads),
//             GT boxes staged to LDS via GLOBAL_LOAD_ASYNC_TO_LDS_B128
//             (ASYNCcnt path), DFL softmax/logsumexp, focal-BCE class sum,
//             32-gt inner loop; single fused LDS tree reduction + atomics.
//   k_final : combine (loss, tb, tc, td) -> d_out[4].
// ---------------------------------------------------------------------------

#define NCLS 80
#define BINS 17
#define GNUM 32
#define TOPKN 10
#define RADIUS 2.5f

typedef __attribute__((ext_vector_type(2))) float v2f;
typedef __attribute__((ext_vector_type(8))) float v8f;

__device__ __forceinline__ float sigmoidf_(float x) {
  return 1.f / (1.f + __expf(-x));
}

// focal_bce(x, t) with alpha=0.25, gamma=2
__device__ __forceinline__ float focal_bce(float x, float t) {
  float p   = sigmoidf_(x);
  float ce  = fmaxf(x, 0.f) - x * t + log1pf(__expf(-fabsf(x)));
  float p_t = p * t + (1.f - p) * (1.f - t);
  float a_t = 0.25f * t + 0.75f * (1.f - t);
  float om  = 1.f - p_t;
  return ce * a_t * om * om;
}

// focal_bce(x, 0): ce = max(x,0)+log1p(e^-|x|); a=0.75; (1-p_t)=p
__device__ __forceinline__ float focal_bce0(float x) {
  float p  = sigmoidf_(x);
  float ce = fmaxf(x, 0.f) + log1pf(__expf(-fabsf(x)));
  return ce * 0.75f * p * p;
}

// ---------------------------------------------------------------------------
__global__ void k_init(float* acc, int* nraw, int n) {
  int t = threadIdx.x;
  if (t < 3) acc[t] = 0.f;
  if (t < n) nraw[t] = 0;
}

// ---------------------------------------------------------------------------
// Count raw positives (in_gt & in_ctr) per image -> nraw[b].
__global__ __launch_bounds__(256) void k_count(
    const float* __restrict__ gtb, int* __restrict__ nraw,
    int W, int HW, float stride) {
  int b = blockIdx.y;
  int p = blockIdx.x * 256 + threadIdx.x;
  int cnt = 0;
  if (p < HW) {
    float cx = (float)(p % W) + 0.5f;
    float cy = (float)(p / W) + 0.5f;
    float inv_s = 1.f / stride;
    for (int g = 0; g < GNUM; ++g) {
      const float* gb = gtb + (size_t)(b * GNUM + g) * 4;
      float x1 = gb[0] * inv_s, y1 = gb[1] * inv_s;
      float x2 = gb[2] * inv_s, y2 = gb[3] * inv_s;
      float dl = cx - x1, dt = cy - y1, dr = x2 - cx, db = y2 - cy;
      bool in_gt = (dl > 0.f) & (dt > 0.f) & (dr > 0.f) & (db > 0.f);
      float gcx = (x1 + x2) * 0.5f, gcy = (y1 + y2) * 0.5f;
      bool in_ctr = (cx >= gcx - RADIUS) & (cx <= gcx + RADIUS) &
                    (cy >= gcy - RADIUS) & (cy <= gcy + RADIUS);
      cnt += (in_gt & in_ctr) ? 1 : 0;
    }
  }
  __shared__ int red[256];
  int t = threadIdx.x;
  red[t] = cnt;
  __syncthreads();
  for (int s = 128; s > 0; s >>= 1) {
    if (t < s) red[t] += red[t + s];
    __syncthreads();
  }
  if (t == 0 && red[0]) atomicAdd(&nraw[b], red[0]);
}

// ---------------------------------------------------------------------------
// Top-10 nearest anchors per (b,gt). One wave per (b, 16-gt tile).
// dist^2 tiles via V_WMMA_F32_16X16X4_F32 (exact: rank-4 bilinear form).
// Unsorted top-10 with replace-max: static indexing only -> stays in VGPRs.
// Anchor grid coords maintained incrementally (step 16 < W): no div in loop.
__global__ __launch_bounds__(32) void k_topk(
    const float* __restrict__ gtb, int* __restrict__ topk_out,
    int W, int HW, float stride) {
  int lane  = threadIdx.x;
  int b     = blockIdx.x;
  int gtile = blockIdx.y;           // 0..1 (GNUM/16)
  int gl    = (gtile << 4) + (lane & 15);

  float inv_s = 1.f / stride;
  const float* gb = gtb + (size_t)(b * GNUM + gl) * 4;
  float gx = (gb[0] + gb[2]) * 0.5f * inv_s;
  float gy = (gb[1] + gb[3]) * 0.5f * inv_s;

  // B matrix (4x16, K x N).  VGPR0: K0 (lanes0-15) / K2 (lanes16-31);
  //                          VGPR1: K1            / K3
  v2f Bm;
  if (lane < 16) { Bm.x = -2.f * gx; Bm.y = -2.f * gy; }
  else           { Bm.x = 1.f;       Bm.y = gx * gx + gy * gy; }

  float best_d[TOPKN];
  int   best_i[TOPKN];
#pragma unroll
  for (int j = 0; j < TOPKN; ++j) { best_d[j] = 3.4e38f; best_i[j] = 0; }
  float worst = 3.4e38f;  // max of best_d
  int   wslot = 0;

  __shared__ float tileD[16][17];   // +1 pad: bank-conflict-free columns

  // incremental grid coordinates for this lane's anchor row (p = t0 + m)
  int xi = lane & 15;               // m < 16 <= W  ->  x=m, y=0
  int yi = 0;

  for (int t0 = 0; t0 < HW; t0 += 16) {
    float cx = (float)xi + 0.5f;
    float cy = (float)yi + 0.5f;
    // A matrix (16x4): VGPR0: K0 / K2 ; VGPR1: K1 / K3
    v2f Am;
    if (lane < 16) { Am.x = cx;                Am.y = cy;  }
    else           { Am.x = cx * cx + cy * cy; Am.y = 1.f; }
    v8f Cz = {};
    v8f D = __builtin_amdgcn_wmma_f32_16x16x4_f32(
        /*neg_a=*/false, Am, /*neg_b=*/false, Bm,
        /*c_mod=*/(short)0, Cz, /*reuse_a=*/false, /*reuse_b=*/false);

    int col   = lane & 15;
    int rbase = (lane >> 4) << 3;   // rows 0..7 / 8..15
#pragma unroll
    for (int r = 0; r < 8; ++r) tileD[rbase + r][col] = D[r];
    __syncthreads();

    if (lane < 16) {
#pragma unroll
      for (int r = 0; r < 16; ++r) {
        float dv = tileD[r][lane];
        if (dv < worst) {            // rare (~10*ln(HW) hits per column)
          int idx = t0 + r;
#pragma unroll
          for (int j = 0; j < TOPKN; ++j)
            if (j == wslot) { best_d[j] = dv; best_i[j] = idx; }
          worst = best_d[0]; wslot = 0;
#pragma unroll
          for (int j = 1; j < TOPKN; ++j)
            if (best_d[j] > worst) { worst = best_d[j]; wslot = j; }
        }
      }
    }
    __syncthreads();

    // advance by one 16-anchor tile; W in {20,40,80} > 16 -> single wrap
    xi += 16;
    if (xi >= W) { xi -= W; ++yi; }
  }

  if (lane < 16) {
    int* outp = topk_out + (size_t)(b * GNUM + gl) * TOPKN;
#pragma unroll
    for (int j = 0; j < TOPKN; ++j) outp[j] = best_i[j];
  }
}

// ---------------------------------------------------------------------------
// Fused main pass: per-anchor DFL softmax + focal class sum + 32-gt loop.
// GT boxes staged to LDS once per block via async global->LDS copy.
__global__ __launch_bounds__(256) void k_main(
    const float* __restrict__ reg, const float* __restrict__ cls,
    const float* __restrict__ gtb, const int* __restrict__ glab,
    const int* __restrict__ topk, const int* __restrict__ nraw,
    float* __restrict__ acc, int W, int HW, float stride) {
  int b = blockIdx.y;
  int p = blockIdx.x * 256 + threadIdx.x;

  __shared__ float gt_s[GNUM * 4];
  __shared__ int   lab_s[GNUM];

  // Stage 32 GT boxes (512B) with GLOBAL_LOAD_ASYNC_TO_LDS_B128: one 16B row
  // per lane of wave 0. Issuing wave waits on ASYNCcnt, then block barrier
  // publishes LDS to all waves.
  if (threadIdx.x < GNUM) {
    unsigned int lds_off =
        (unsigned int)(unsigned long long)(&gt_s[threadIdx.x * 4]);
    const float* src = gtb + (size_t)(b * GNUM + threadIdx.x) * 4;
    asm volatile("global_load_async_to_lds_b128 %0, %1, off"
                 :
                 : "v"(lds_off), "v"(src)
                 : "memory");
    lab_s[threadIdx.x] = glab[b * GNUM + threadIdx.x];
  }
  asm volatile("s_wait_asynccnt 0" ::: "memory");
  __syncthreads();

  float box_acc = 0.f, cls_acc = 0.f, dfl_acc = 0.f;

  int  nr      = nraw[b];
  bool any_pos = nr > 0;
  float inv_np4 = 1.f / ((any_pos ? (float)nr : (float)(GNUM * TOPKN)) * 4.f);

  if (p < HW) {
    float cx = (float)(p % W) + 0.5f;
    float cy = (float)(p / W) + 0.5f;

    const float* regb = reg + (size_t)b * (4 * BINS) * HW + p;
    const float* clsb = cls + (size_t)b * NCLS * HW + p;

    // ---- DFL softmax per side k: logsumexp + expectation -------------
    float lse[4], dexp[4];
    for (int k = 0; k < 4; ++k) {
      __builtin_prefetch(regb + (size_t)((k + 1) * BINS) * HW, 0, 1);
      float v[BINS];
      float mx = -3.4e38f;
#pragma unroll
      for (int j = 0; j < BINS; ++j) {
        v[j] = regb[(size_t)(k * BINS + j) * HW];
        mx = fmaxf(mx, v[j]);
      }
      float s = 0.f, num = 0.f;
#pragma unroll
      for (int j = 0; j < BINS; ++j) {
        float e = __expf(v[j] - mx);
        s += e;
        num += e * (float)j;
      }
      lse[k]  = mx + __logf(s);
      dexp[k] = num / s;
    }
    float px1 = cx - dexp[0], py1 = cy - dexp[1];
    float px2 = cx + dexp[2], py2 = cy + dexp[3];
    float a1 = (px2 - px1) * (py2 - py1);

    // ---- focal-BCE(neg) sum over 80 classes --------------------------
    float fz_sum = 0.f;
    for (int c = 0; c < NCLS; ++c) {
      if ((c & 15) == 0 && c + 16 < NCLS)
        __builtin_prefetch(clsb + (size_t)(c + 16) * HW, 0, 1);
      fz_sum += focal_bce0(clsb[(size_t)c * HW]);
    }

    // ---- per-gt loop (GT data from LDS) ------------------------------
    bool pos_pt = false;
    float inv_s = 1.f / stride;
    for (int g = 0; g < GNUM; ++g) {
      float x1 = gt_s[g * 4 + 0] * inv_s, y1 = gt_s[g * 4 + 1] * inv_s;
      float x2 = gt_s[g * 4 + 2] * inv_s, y2 = gt_s[g * 4 + 3] * inv_s;
      float dl = cx - x1, dt = cy - y1, dr = x2 - cx, db = y2 - cy;

      bool m;
      if (any_pos) {
        bool in_gt = (dl > 0.f) & (dt > 0.f) & (dr > 0.f) & (db > 0.f);
        float gcx = (x1 + x2) * 0.5f, gcy = (y1 + y2) * 0.5f;
        bool in_ctr = (cx >= gcx - RADIUS) & (cx <= gcx + RADIUS) &
                      (cy >= gcy - RADIUS) & (cy <= gcy + RADIUS);
        m = in_gt & in_ctr;
      } else {
        m = false;
        const int* tk = topk + (size_t)(b * GNUM + g) * TOPKN;
#pragma unroll
        for (int j = 0; j < TOPKN; ++j) m = m | (tk[j] == p);
      }
      if (!m) continue;
      pos_pt = true;

      // IoU (pred vs gt)
      float ix1 = fmaxf(px1, x1), iy1 = fmaxf(py1, y1);
      float ix2 = fminf(px2, x2), iy2 = fminf(py2, y2);
      float iw = fmaxf(ix2 - ix1, 0.f), ih = fmaxf(iy2 - iy1, 0.f);
      float inter = iw * ih;
      float a2 = (x2 - x1) * (y2 - y1);
      float iou = inter / (a1 + a2 - inter + 1e-6f);
      box_acc += 1.f - iou;

      // classification positive term
      int lab = lab_s[g];
      float xat = clsb[(size_t)lab * HW];
      cls_acc += fz_sum - focal_bce0(xat) + focal_bce(xat, iou);

      // DFL pair term
      float lt[4] = {dl, dt, dr, db};
      float dsum = 0.f;
#pragma unroll
      for (int k = 0; k < 4; ++k) {
        float tgt = fminf(fmaxf(lt[k], 0.f), 15.9999f);
        int lb = (int)floorf(tgt);
        int rb = lb + 1;
        if (rb > 16) rb = 16;
        float wl = (float)rb - tgt;
        float wr = tgt - (float)lb;
        float lp_l = regb[(size_t)(k * BINS + lb) * HW] - lse[k];
        float lp_r = regb[(size_t)(k * BINS + rb) * HW] - lse[k];
        dsum -= lp_l * wl + lp_r * wr;
      }
      dfl_acc += dsum * inv_np4;
    }
    if (!pos_pt) cls_acc += fz_sum;  // negative-point classification term
  }

  // ---- fused block reduction (one LDS tree for all 3 accumulators) ----
  __shared__ float redA[256], redB[256], redC[256];
  int t = threadIdx.x;
  redA[t] = box_acc;
  redB[t] = cls_acc;
  redC[t] = dfl_acc;
  __syncthreads();
  for (int s = 128; s > 0; s >>= 1) {
    if (t < s) {
      redA[t] += redA[t + s];
      redB[t] += redB[t + s];
      redC[t] += redC[t + s];
    }
    __syncthreads();
  }
  if (t == 0) {
    atomicAdd(&acc[0], redA[0]);
    atomicAdd(&acc[1], redB[0]);
    atomicAdd(&acc[2], redC[0]);
  }
}

// ---------------------------------------------------------------------------
__global__ void k_final(const float* __restrict__ acc, float* __restrict__ out) {
  if (threadIdx.x == 0) {
    float tb = acc[0], tc = acc[1], td = acc[2];
    out[0] = 7.5f * tb + 1.0f * tc + 1.5f * td;
    out[1] = tb;
    out[2] = tc;
    out[3] = td;
  }
}

// ---------------------------------------------------------------------------
extern "C" void kernel_launch(void* const* d_in, const int* in_sizes, int n_in,
                              void* d_out, int out_size, void* d_ws, size_t ws_size,
                              hipStream_t stream) {
  const float* reg[3] = {(const float*)d_in[0], (const float*)d_in[2],
                         (const float*)d_in[4]};
  const float* cls[3] = {(const float*)d_in[1], (const float*)d_in[3],
                         (const float*)d_in[5]};
  const float* gtb  = (const float*)d_in[6];
  const int*   glab = (const int*)d_in[7];

  int Bn = in_sizes[7] / GNUM;   // 16
  int HW[3], Wd[3];
  float strd[3] = {8.f, 16.f, 32.f};
  for (int l = 0; l < 3; ++l) {
    HW[l] = in_sizes[2 * l] / (Bn * 4 * BINS);
    int w = 1;
    while ((w + 1) * (w + 1) <= HW[l]) ++w;   // integer sqrt (square levels)
    Wd[l] = w;
  }

  // Workspace layout: [0..12) acc(3 floats) | 64: nraw[3][Bn] | 4096: topk
  char*  ws   = (char*)d_ws;
  float* acc  = (float*)ws;
  int*   nraw = (int*)(ws + 64);
  int*   topk = (int*)(ws + 4096);

  hipLaunchKernelGGL(k_init, dim3(1), dim3(256), 0, stream, acc, nraw, 3 * Bn);

  for (int l = 0; l < 3; ++l) {
    dim3 grid((HW[l] + 255) / 256, Bn);
    hipLaunchKernelGGL(k_count, grid, dim3(256), 0, stream,
                       gtb, nraw + l * Bn, Wd[l], HW[l], strd[l]);
  }
  for (int l = 0; l < 3; ++l) {
    hipLaunchKernelGGL(k_topk, dim3(Bn, GNUM / 16), dim3(32), 0, stream,
                       gtb, topk + (size_t)l * Bn * GNUM * TOPKN,
                       Wd[l], HW[l], strd[l]);
  }
  for (int l = 0; l < 3; ++l) {
    dim3 grid((HW[l] + 255) / 256, Bn);
    hipLaunchKernelGGL(k_main, grid, dim3(256), 0, stream,
                       reg[l], cls[l], gtb, glab,
                       topk + (size_t)l * Bn * GNUM * TOPKN,
                       nraw + l * Bn, acc, Wd[l], HW[l], strd[l]);
  }
  hipLaunchKernelGGL(k_final, dim3(1), dim3(32), 0, stream, acc, (float*)d_out);
}